// STLocalizedConv_9663676416327
// MI455X (gfx1250) — compile-verified
//
#include <hip/hip_runtime.h>

#define B_  16
#define T_  12
#define N_  512
#define D_  96
#define G_  4
#define KT_ 3
#define S_  10            // T - KT + 1
#define KD_ 288           // KT*D
#define GD_ 480           // D*(G+1)
#define M1_ (B_*S_*N_)    // 81920

typedef float v2f __attribute__((ext_vector_type(2)));
typedef float v8f __attribute__((ext_vector_type(8)));

__device__ __forceinline__ v8f wmma4(v2f a, v2f b, v8f c) {
  // D = A(16x4,f32) * B(4x16,f32) + C(16x16,f32)
  return __builtin_amdgcn_wmma_f32_16x16x4_f32(
      /*neg_a=*/false, a, /*neg_b=*/false, b,
      /*c_mod=*/(short)0, c, /*reuse_a=*/false, /*reuse_b=*/false);
}

// ---------------------------------------------------------------------------
// Kernel 1: h = relu(Xw @ W_updt^T), h stored flat as (B*S*N, 288)
//   Xw[m, kt*96+d] = X[b, s+kt, n, d]  with m = (b*S+s)*N + n
// One 16x16 output tile per wave; K-loop = 288/4 = 72 WMMAs.
// ---------------------------------------------------------------------------
__global__ void k1_updt(const float* __restrict__ X,
                        const float* __restrict__ Wu,
                        float* __restrict__ h) {
  const int wave = threadIdx.x >> 5;
  const int lane = threadIdx.x & 31;
  const int half = lane >> 4;
  const int lid  = lane & 15;

  const int tile = blockIdx.x * 8 + wave;     // 0 .. 92159
  const int it   = tile % (KD_ / 16);         // 0..17  (i-tile)
  const int mt   = tile / (KD_ / 16);         // 0..5119 (m-tile)
  const int m0   = mt * 16;

  // per-lane A row (rows m0..m0+15 share b,s since N%16==0)
  const int m  = m0 + lid;
  const int bs = m / N_;
  const int n  = m % N_;
  const int b  = bs / S_;
  const int s  = bs % S_;

  const float* pA[3];
#pragma unroll
  for (int kt = 0; kt < KT_; ++kt)
    pA[kt] = X + ((size_t)((b * T_ + s + kt) * N_) + n) * D_ + 2 * half;

  // B[k][col] = W_updt[col][k]  (row-major 288 stride)
  const int col = it * 16 + lid;
  const float* pB = Wu + (size_t)col * KD_ + 2 * half;

  v8f c = {};
#pragma unroll
  for (int kt = 0; kt < KT_; ++kt) {
    const float* pa = pA[kt];
    const float* pb = pB + kt * D_;
    for (int kk = 0; kk < D_; kk += 4) {
      v2f a  = *(const v2f*)(pa + kk);
      v2f bb = *(const v2f*)(pb + kk);
      c = wmma4(a, bb, c);
    }
  }

  // ReLU + store: element r -> row m0 + r + 8*half, col it*16+lid
  float* ho = h + (size_t)m0 * KD_ + it * 16 + lid;
#pragma unroll
  for (int r = 0; r < 8; ++r) {
    float v = c[r];
    ho[(size_t)(r + 8 * half) * KD_] = v > 0.0f ? v : 0.0f;
  }
}

// ---------------------------------------------------------------------------
// Kernel 2 (fused), 192 threads = 6 waves per block, one block per
// (b, s, 16-row slab of n):
//   Phase A: X0 = mean_kt(h)                       -> LDS cols [0,96)
//   Phase B: Hk_g = dyn[g,b] @ Xk[b,s]             -> LDS cols [96,480)
//            wave w owns column tile d0 = 16*w and accumulates ALL 4 g's
//            at once: each strided Xk (B-operand) load pair feeds 4 WMMAs,
//            cutting the worst-pattern loads 4x vs. one-tile-per-wave.
//   Phase C: out = Hcat @ W_gcn^T + b_gcn  (16 x 96), wave w -> cols 16*w
// Hk never touches HBM.
// ---------------------------------------------------------------------------
__global__ void k2_gcn(const float* __restrict__ dyn,
                       const float* __restrict__ h,
                       const float* __restrict__ Wg,
                       const float* __restrict__ bg,
                       float* __restrict__ out) {
  __shared__ float Hcat[16][GD_ + 4];   // 16 x 484 f32, +4 pad

  const int wave = threadIdx.x >> 5;    // 0..5
  const int lane = threadIdx.x & 31;
  const int half = lane >> 4;
  const int lid  = lane & 15;

  const int ntile = blockIdx.x & 31;    // N_/16 = 32
  const int bs    = blockIdx.x >> 5;    // 0 .. B*S-1
  const int b     = bs / S_;
  const int n0    = ntile * 16;

  // ---- Phase A: X0 (mean over kt), all 192 threads ----
  for (int e = threadIdx.x; e < 16 * D_; e += 192) {
    const int r = e / D_, d = e % D_;
    const float* p = h + (size_t)(bs * N_ + n0 + r) * KD_ + d;
    Hcat[r][d] = (p[0] + p[D_] + p[2 * D_]) * (1.0f / 3.0f);
  }

  // ---- Phase B: wave = d-tile, 4 g-accumulators, K = 1536 ----
  {
    const int d0 = wave * 16;

    // A rows: dyn_graph[g, b, n0+lid, :]  (stride KT_*N_ = 1536)
    const float* pA[G_];
#pragma unroll
    for (int g = 0; g < G_; ++g)
      pA[g] = dyn + ((size_t)((g * B_ + b) * N_) + n0 + lid) * (KT_ * N_) +
              2 * half;

    v8f acc[G_];
#pragma unroll
    for (int g = 0; g < G_; ++g) acc[g] = (v8f){};

#pragma unroll
    for (int kt = 0; kt < KT_; ++kt) {
      // B col: Xk[b,s][m][d0+lid], m = kt*512+n'
      //   -> h[(bs*1536+kt)*96 + d0+lid + n'*288]
      const float* pb = h + (size_t)(bs * (KT_ * N_) + kt) * D_ + d0 + lid;
      const int ktoff = kt * N_;
      for (int nn = 0; nn < N_; nn += 4) {
        const int mrow = nn + 2 * half;
        v2f bb;
        bb.x = pb[(size_t)mrow * KD_];
        bb.y = pb[(size_t)(mrow + 1) * KD_];
#pragma unroll
        for (int g = 0; g < G_; ++g) {
          v2f a = *(const v2f*)(pA[g] + ktoff + nn);
          acc[g] = wmma4(a, bb, acc[g]);
        }
      }
    }

#pragma unroll
    for (int g = 0; g < G_; ++g)
#pragma unroll
      for (int r = 0; r < 8; ++r)
        Hcat[r + 8 * half][D_ + g * D_ + d0 + lid] = acc[g][r];
  }

  __syncthreads();

  // ---- Phase C: out(16x96) = Hcat(16x480) @ W_gcn^T + bias ----
  {
    const int d0 = wave * 16;
    const float* pB = Wg + (size_t)(d0 + lid) * GD_ + 2 * half;
    v8f c = {};
    for (int k0 = 0; k0 < GD_; k0 += 4) {
      v2f a  = *(const v2f*)(&Hcat[lid][k0 + 2 * half]);
      v2f bb = *(const v2f*)(pB + k0);
      c = wmma4(a, bb, c);
    }
    const float bias = bg[d0 + lid];
    float* po = out + (size_t)(bs * N_ + n0) * D_ + d0 + lid;
#pragma unroll
    for (int r = 0; r < 8; ++r)
      po[(size_t)(r + 8 * half) * D_] = c[r] + bias;
  }
}

// ---------------------------------------------------------------------------
extern "C" void kernel_launch(void* const* d_in, const int* in_sizes, int n_in,
                              void* d_out, int out_size, void* d_ws, size_t ws_size,
                              hipStream_t stream) {
  const float* X   = (const float*)d_in[0];  // (B,T,N,D)
  const float* dyn = (const float*)d_in[1];  // (G,B,N,KT*N)
  const float* Wu  = (const float*)d_in[2];  // (288,288)
  const float* Wg  = (const float*)d_in[3];  // (96,480)
  const float* bg  = (const float*)d_in[4];  // (96,)
  float* out = (float*)d_out;                // (B,S,N,D)
  float* h   = (float*)d_ws;                 // (B*S*N, 288) f32 = 94.4 MB

  // Kernel 1: 5120 m-tiles * 18 i-tiles = 92160 wave-tiles / 8 waves per block
  k1_updt<<<(M1_ / 16) * (KD_ / 16) / 8, 256, 0, stream>>>(X, Wu, h);
  // Kernel 2: one block (6 waves) per (b, s, 16-row slab of n)
  k2_gcn<<<B_ * S_ * (N_ / 16), 192, 0, stream>>>(dyn, h, Wg, bg, out);
}